// Attention_20040317403762
// MI455X (gfx1250) — compile-verified
//
#include <hip/hip_runtime.h>
#include <hip/hip_bf16.h>
#include <math.h>

// Attention with single-query token, algebraically refactored:
//   qp  = query @ Wq                       (16x1024, WMMA f32)
//   qk[b] = Wk @ qp[b]                     (16x1024, WMMA f32)
//   scores[b,s] = key[b,s] . qk[b] / 32    (streams 256MB, memory bound)
//   attn = softmax(scores)
//   av[b] = attn[b] @ value[b]             (streams 256MB, memory bound)
//   out  = av @ Wv                         (16x1024, WMMA f32)
// Memory floor: ~524MB / 23.3TB/s ~= 22us. FLOPs reduced 275G -> ~0.4G.

typedef float v2f __attribute__((ext_vector_type(2)));
typedef float v8f __attribute__((ext_vector_type(8)));

#define DIM 1024
#define ISL 4096
#define NB  16
#define SCHUNKS 32
#define SLEN (ISL / SCHUNKS)   // 128

// ---------------------------------------------------------------------------
// Wave-level 16xN GEMM tile: D[0:16, n0:n0+16] = A[0:16, 0:K] * B[0:K, n0:n0+16]
// A,B,D row-major f32. One wave (32 lanes), V_WMMA_F32_16X16X4_F32 over K.
// A frag (16x4): lanes 0-15 M=lane {K=k0,k0+1}; lanes 16-31 M=lane-16 {K=k0+2,k0+3}
// B frag (4x16): lanes 0-15 N=lane rows {k0,k0+1}; lanes 16-31 rows {k0+2,k0+3}
// D (16x16): vgpr r -> row r (lanes 0-15) / row 8+r (lanes 16-31), col = lane&15
// ---------------------------------------------------------------------------
__device__ __forceinline__ void wave_gemm16(const float* __restrict__ A, int lda,
                                            const float* __restrict__ B, int ldb,
                                            float* __restrict__ D, int ldd,
                                            int n0, int K, int lane) {
  const int l  = lane & 15;
  const int kk = (lane >> 4) << 1;  // 0 or 2
  v8f acc = {};
  for (int k0 = 0; k0 < K; k0 += 4) {
    v2f a, b;
    a.x = A[l * lda + k0 + kk];
    a.y = A[l * lda + k0 + kk + 1];
    b.x = B[(k0 + kk) * ldb + n0 + l];
    b.y = B[(k0 + kk + 1) * ldb + n0 + l];
    acc = __builtin_amdgcn_wmma_f32_16x16x4_f32(false, a, false, b,
                                                (short)0, acc, false, false);
  }
  const int rbase = (lane < 16) ? 0 : 8;
#pragma unroll
  for (int r = 0; r < 8; ++r)
    D[(rbase + r) * ldd + n0 + l] = acc[r];
}

// qp = query(16x1024) @ Wq(1024x1024); one wave per 16-wide N tile (64 blocks)
__global__ __launch_bounds__(32) void qproj_kernel(const float* __restrict__ query,
                                                   const float* __restrict__ Wq,
                                                   float* __restrict__ qp) {
  wave_gemm16(query, DIM, Wq, DIM, qp, DIM, blockIdx.x * 16, DIM, threadIdx.x);
}

// out = av(16x1024) @ Wv(1024x1024)
__global__ __launch_bounds__(32) void out_kernel(const float* __restrict__ av,
                                                 const float* __restrict__ Wv,
                                                 float* __restrict__ out) {
  wave_gemm16(av, DIM, Wv, DIM, out, DIM, blockIdx.x * 16, DIM, threadIdx.x);
}

// qk[b][i] = sum_h Wk[i][h] * qp[b][h]   (M=1024 rows of Wk, N=16 batches)
// A = Wk tile, B[k][n] = qp[n][k] (transposed read), D stored transposed to qk[b][i]
__global__ __launch_bounds__(32) void qk_kernel(const float* __restrict__ Wk,
                                                const float* __restrict__ qp,
                                                float* __restrict__ qk) {
  const int lane = threadIdx.x;
  const int i0   = blockIdx.x * 16;     // row tile over 1024
  const int l    = lane & 15;
  const int kk   = (lane >> 4) << 1;
  v8f acc = {};
  for (int k0 = 0; k0 < DIM; k0 += 4) {
    v2f a, b;
    a.x = Wk[(i0 + l) * DIM + k0 + kk];
    a.y = Wk[(i0 + l) * DIM + k0 + kk + 1];
    b.x = qp[l * DIM + k0 + kk];          // B[k][n] = qp[n][k], n = l
    b.y = qp[l * DIM + k0 + kk + 1];
    acc = __builtin_amdgcn_wmma_f32_16x16x4_f32(false, a, false, b,
                                                (short)0, acc, false, false);
  }
  const int rbase = (lane < 16) ? 0 : 8;
#pragma unroll
  for (int r = 0; r < 8; ++r)
    qk[l * DIM + i0 + rbase + r] = acc[r];  // qk[b=l][i]
}

// scores[b][s] = key[b][s] . qk[b] * (1/32). One wave per 4KB row; 8 rows/block.
__global__ __launch_bounds__(256) void scores_kernel(const float* __restrict__ key,
                                                     const float* __restrict__ qk,
                                                     float* __restrict__ scores) {
  __shared__ float qks[DIM];
  const int tid = threadIdx.x;
  const int b   = blockIdx.x >> 9;            // 512 blocks per batch
  const int s0  = (blockIdx.x & 511) << 3;    // 8 rows per block
  ((float4*)qks)[tid] = ((const float4*)(qk + b * DIM))[tid];  // 4KB -> LDS
  __syncthreads();
  const int wave = tid >> 5, lane = tid & 31;
  const int s = s0 + wave;
  const float4* k4 = (const float4*)(key + (size_t)(b * ISL + s) * DIM);
  const float4* q4 = (const float4*)qks;
  float acc = 0.0f;
#pragma unroll
  for (int w = 0; w < 8; ++w) {
    float4 kv = k4[w * 32 + lane];
    float4 qv = q4[w * 32 + lane];
    acc = fmaf(kv.x, qv.x, acc);
    acc = fmaf(kv.y, qv.y, acc);
    acc = fmaf(kv.z, qv.z, acc);
    acc = fmaf(kv.w, qv.w, acc);
  }
#pragma unroll
  for (int off = 16; off > 0; off >>= 1)
    acc += __shfl_xor(acc, off, 32);
  if (lane == 0)
    scores[b * ISL + s] = acc * 0.03125f;     // 1/sqrt(1024)
}

// In-place softmax over 4096 per batch; one block per batch.
__global__ __launch_bounds__(256) void softmax_kernel(float* __restrict__ sc) {
  __shared__ float red[256];
  const int tid = threadIdx.x;
  float* s = sc + blockIdx.x * ISL;
  float m = -3.402823466e38f;
  for (int i = tid; i < ISL; i += 256) m = fmaxf(m, s[i]);
  red[tid] = m; __syncthreads();
  for (int off = 128; off > 0; off >>= 1) {
    if (tid < off) red[tid] = fmaxf(red[tid], red[tid + off]);
    __syncthreads();
  }
  m = red[0]; __syncthreads();
  float sum = 0.0f;
  for (int i = tid; i < ISL; i += 256) {
    float e = __expf(s[i] - m);
    s[i] = e;
    sum += e;
  }
  red[tid] = sum; __syncthreads();
  for (int off = 128; off > 0; off >>= 1) {
    if (tid < off) red[tid] += red[tid + off];
    __syncthreads();
  }
  const float inv = 1.0f / red[0];
  for (int i = tid; i < ISL; i += 256) s[i] *= inv;
}

// av_part[sch][b][c] = sum_{s in chunk} attn[b][s] * value[b][s][c]
// float4 per lane (b128 loads), prefetch ~256KB ahead. 512 blocks x 256 thr.
__global__ __launch_bounds__(256) void av_partial_kernel(const float* __restrict__ value,
                                                         const float* __restrict__ attn,
                                                         float* __restrict__ av_part) {
  __shared__ float a_s[SLEN];
  const int tid   = threadIdx.x;
  const int b     = blockIdx.x >> 5;          // 32 s-chunks per batch
  const int sch   = blockIdx.x & 31;
  const int sbase = sch * SLEN;
  if (tid < SLEN) a_s[tid] = attn[b * ISL + sbase + tid];
  __syncthreads();
  const float4* v4 = (const float4*)(value + (size_t)(b * ISL + sbase) * DIM);
  float4 acc = make_float4(0.f, 0.f, 0.f, 0.f);
  for (int s0 = 0; s0 < SLEN; s0 += 8) {
    const int ps = (s0 + 64 < SLEN) ? (s0 + 64) : (SLEN - 1);
    __builtin_prefetch(&v4[(size_t)ps * 256 + tid], 0, 1);   // global_prefetch_b8
#pragma unroll
    for (int u = 0; u < 8; ++u) {
      const float  a = a_s[s0 + u];
      const float4 v = v4[(size_t)(s0 + u) * 256 + tid];
      acc.x = fmaf(a, v.x, acc.x);
      acc.y = fmaf(a, v.y, acc.y);
      acc.z = fmaf(a, v.z, acc.z);
      acc.w = fmaf(a, v.w, acc.w);
    }
  }
  ((float4*)av_part)[(size_t)(sch * NB + b) * 256 + tid] = acc;
}

// av[b][c] = sum over 32 partials
__global__ __launch_bounds__(256) void av_reduce_kernel(const float* __restrict__ av_part,
                                                        float* __restrict__ av) {
  const int idx = blockIdx.x * 256 + threadIdx.x;   // 0..16383
  float acc = 0.0f;
#pragma unroll
  for (int p = 0; p < SCHUNKS; ++p)
    acc += av_part[p * (NB * DIM) + idx];
  av[idx] = acc;
}

extern "C" void kernel_launch(void* const* d_in, const int* in_sizes, int n_in,
                              void* d_out, int out_size, void* d_ws, size_t ws_size,
                              hipStream_t stream) {
  (void)in_sizes; (void)n_in; (void)out_size; (void)ws_size;
  const float* key   = (const float*)d_in[0];   // 16 x 4096 x 1024
  const float* query = (const float*)d_in[1];   // 16 x 1 x 1024
  const float* value = (const float*)d_in[2];   // 16 x 4096 x 1024
  const float* Wk    = (const float*)d_in[3];   // 1024 x 1024
  const float* Wq    = (const float*)d_in[4];   // 1024 x 1024
  const float* Wv    = (const float*)d_in[5];   // 1024 x 1024
  float* out = (float*)d_out;                   // 16 x 1024

  char* ws = (char*)d_ws;
  float* qp      = (float*)(ws);                         //  64 KB: 16x1024
  float* qk      = (float*)(ws + (64u << 10));           //  64 KB: 16x1024
  float* scores  = (float*)(ws + (128u << 10));          // 256 KB: 16x4096 (becomes attn)
  float* av_part = (float*)(ws + (384u << 10));          //   2 MB: 32x16x1024
  float* av      = (float*)(ws + (384u << 10) + (SCHUNKS * NB * DIM * 4u)); // 64 KB

  qproj_kernel     <<<DIM / 16, 32,  0, stream>>>(query, Wq, qp);
  qk_kernel        <<<DIM / 16, 32,  0, stream>>>(Wk, qp, qk);
  scores_kernel    <<<NB * (ISL / 8), 256, 0, stream>>>(key, qk, scores);
  softmax_kernel   <<<NB, 256, 0, stream>>>(scores);
  av_partial_kernel<<<NB * SCHUNKS, 256, 0, stream>>>(value, scores, av_part);
  av_reduce_kernel <<<(NB * DIM) / 256, 256, 0, stream>>>(av_part, av);
  out_kernel       <<<DIM / 16, 32,  0, stream>>>(av, Wv, out);
}